// LinearModel_43250320670751
// MI455X (gfx1250) — compile-verified
//
#include <hip/hip_runtime.h>
#include <hip/hip_bf16.h>

// Problem constants (from reference):
//   x: int32 [256, 8] token ids in [0, 2048)
//   W: fp32  [16384, 16384]  (out_features x in_features)
//   b: fp32  [16384]
//   out[b, r] = bias[r] + sum_{j=0..7} W[r, j*2048 + x[b,j]]   -> [256, 16384]
#define BATCH     256
#define CTX       8
#define VOCAB     2048
#define DTOT      16384
#define RTILE     16          // output rows per workgroup
#define NTHREADS  256         // 8 waves (wave32), thread == batch row
#define CHUNKS_PER_STAGE ((RTILE * VOCAB / 4) / NTHREADS)   // 32 B128 ops/thread/stage

// CDNA5 async global->LDS path (ASYNCcnt-tracked DMA), with sync fallback.
#if defined(__HIP_DEVICE_COMPILE__) && __has_builtin(__builtin_amdgcn_global_load_async_to_lds_b128)
#define USE_ASYNC 1
#else
#define USE_ASYNC 0
#if defined(__HIP_DEVICE_COMPILE__)
#warning "gfx1250 async-to-LDS builtin not found: using sync global->LDS fallback"
#endif
#endif

// Pointee type the builtin expects: GCC-style 16-byte int vector.
typedef int v4i __attribute__((vector_size(16)));
typedef __attribute__((address_space(1))) v4i gv4i_t;   // global (prints as __device__)
typedef __attribute__((address_space(3))) v4i lv4i_t;   // LDS    (prints as __shared__)

// Wait until this wave's outstanding async-to-LDS count <= N (in-order completion,
// so N==CHUNKS_PER_STAGE means "all but the newest stage have landed").
template <int N>
__device__ __forceinline__ void wait_async_le() {
#if USE_ASYNC
#if __has_builtin(__builtin_amdgcn_s_wait_asynccnt)
  __builtin_amdgcn_s_wait_asynccnt(N);
#else
  asm volatile("s_wait_asynccnt %0" ::"i"(N) : "memory");
#endif
#endif
}

// Stage RTILE rows x VOCAB cols of W (row stride DTOT floats) into contiguous LDS.
// 128 KB per stage; per thread: 32 x async B128 transfers, fully coalesced.
__device__ __forceinline__ void stage_segment(const float* __restrict__ gbase,
                                              float* __restrict__ lbuf, int tid) {
#pragma unroll
  for (int k = 0; k < CHUNKS_PER_STAGE; ++k) {
    int c    = k * NTHREADS + tid;   // float4 chunk id in [0, 8192)
    int row  = c >> 9;               // 512 float4-chunks per 2048-wide row
    int col4 = c & 511;
    const float* g = gbase + (size_t)row * DTOT + (size_t)col4 * 4;
    float*       l = lbuf + ((row << 11) + (col4 << 2));
#if USE_ASYNC
    float* gnc = const_cast<float*>(g);
    __builtin_amdgcn_global_load_async_to_lds_b128(
        (gv4i_t*)gnc, (lv4i_t*)l, /*offset=*/0, /*cpol=*/0);
#else
    *(float4*)l = *(const float4*)g;
#endif
  }
}

extern "C" __global__ void __launch_bounds__(NTHREADS, 1)
gather_sum_rows_kernel(const int* __restrict__ x, const float* __restrict__ W,
                       const float* __restrict__ bias, float* __restrict__ out) {
  extern __shared__ float smem[];
  float* buf[2] = {smem, smem + RTILE * VOCAB};  // 2 x 128 KB double buffer
  float* bias_s = smem + 2 * RTILE * VOCAB;      // 16 floats

  const int tid = threadIdx.x;              // == batch row b
  const int r0  = blockIdx.x * RTILE;
  const float* wt = W + (size_t)r0 * DTOT;  // this tile's row block

  // Per-thread token indices for its batch row (values in [0, VOCAB)).
  int xv[CTX];
#pragma unroll
  for (int j = 0; j < CTX; ++j) xv[j] = x[tid * CTX + j];

  if (tid < RTILE) bias_s[tid] = bias[r0 + tid];

  // 2-deep async pipeline: pre-issue stages 0 and 1 (up to 256 KB in flight/WGP).
  stage_segment(wt, buf[0], tid);
  stage_segment(wt + (size_t)VOCAB, buf[1], tid);

  float acc[RTILE];
#pragma unroll
  for (int rr = 0; rr < RTILE; ++rr) acc[rr] = 0.0f;

#pragma unroll
  for (int j = 0; j < CTX; ++j) {
    // Stage j landed when at most the newest stage (j+1) is still outstanding.
    if (j < CTX - 1) wait_async_le<CHUNKS_PER_STAGE>();
    else             wait_async_le<0>();
    __syncthreads();                     // all waves' DMAs for stage j visible

    const int c = xv[j];
#pragma unroll
    for (int rr = 0; rr < RTILE; ++rr)
      acc[rr] += buf[j & 1][rr * VOCAB + c];   // random-index LDS gather

    __syncthreads();                     // all waves done reading buf[j&1]
    if (j + 2 < CTX)                     // refill freed buffer with stage j+2
      stage_segment(wt + (size_t)(j + 2) * VOCAB, buf[j & 1], tid);
  }

  // Transpose accumulators through LDS (pad stride 17 -> conflict-free) so
  // global stores are 64B-contiguous runs instead of 64KB-strided scalars.
  float* trans = buf[0];  // safe: all DMAs drained, all reads barriered
#pragma unroll
  for (int rr = 0; rr < RTILE; ++rr) trans[tid * (RTILE + 1) + rr] = acc[rr];
  __syncthreads();

#pragma unroll
  for (int i = 0; i < RTILE; ++i) {
    int e  = i * NTHREADS + tid;     // element in [0, 256*16)
    int b  = e >> 4;                 // batch
    int rr = e & (RTILE - 1);        // row within tile
    out[(size_t)b * DTOT + r0 + rr] = trans[b * (RTILE + 1) + rr] + bias_s[rr];
  }
}

extern "C" void kernel_launch(void* const* d_in, const int* in_sizes, int n_in,
                              void* d_out, int out_size, void* d_ws, size_t ws_size,
                              hipStream_t stream) {
  const int*   x    = (const int*)d_in[0];    // [256*8]
  const float* W    = (const float*)d_in[1];  // [16384*16384]
  const float* bias = (const float*)d_in[2];  // [16384]
  float*       out  = (float*)d_out;          // [256*16384]

  const size_t shmem = (size_t)(2 * RTILE * VOCAB + RTILE) * sizeof(float); // 262,208 B
  dim3 grid(DTOT / RTILE);  // 1024 workgroups
  dim3 block(NTHREADS);     // 8 wave32 waves
  gather_sum_rows_kernel<<<grid, block, shmem, stream>>>(x, W, bias, out);
}